// ODEFunc_81570018885914
// MI455X (gfx1250) — compile-verified
//
#include <hip/hip_runtime.h>

typedef __attribute__((ext_vector_type(2))) float v2f;
typedef __attribute__((ext_vector_type(8))) float v8f;

#define NUM_VAR 20
#define HIDDEN  50
#define LDS_STRIDE 52          // floats; even (8B rows) and conflict-free for our patterns
#define WAVES_PER_BLOCK 8

// One wave processes 16 rows per iteration:
//   GEMM1: H(16x64) = A(16x4: x=y^3, K=2 padded) x B_nt(4x16: W1_eff^T col tiles), 4 WMMAs
//   tanhshrink (hw v_tanh_f32) -> stage 16x50 tile in private LDS (stride 52)
//   GEMM2: OUT^T(2x16) = W2(16x4 tiles, rows>=2 zero) x A^T(4x16 from LDS), 13 WMMAs
__global__ __launch_bounds__(256, 1) void odefunc_mlp_kernel(
    const float* __restrict__ y,
    const float* __restrict__ w1,
    const float* __restrict__ b1,
    const float* __restrict__ w2,
    const float* __restrict__ b2,
    const float* __restrict__ sp,
    float* __restrict__ out,
    int n_rows)
{
  __shared__ float lds[WAVES_PER_BLOCK][16 * LDS_STRIDE];

  const int lane = threadIdx.x & 31;
  const int wave = threadIdx.x >> 5;
  const int lid  = lane & 15;   // N / M index within WMMA tiles
  const int half = lane >> 4;   // K-half selector

  float* Lp = lds[wave];

  // ---------------- loop-invariant WMMA operands ----------------
  // GEMM1 B tiles: B_nt[k][n] = W1_eff[n+16*nt][k], k in {0,1}; k=2,3 rows are zero.
  // f32 4x16 B layout: lanes 0-15 hold K=0(v0),K=1(v1); lanes 16-31 hold K=2,K=3.
  v2f  bw[4];
  float b1v[4];
#pragma unroll
  for (int nt = 0; nt < 4; ++nt) {
    const int col = lid + 16 * nt;
    float e0 = 0.f, e1 = 0.f;
    if (half == 0 && col < HIDDEN) {
      const float* wrow = (col < NUM_VAR) ? (sp + 2 * col) : (w1 + 2 * col);
      e0 = wrow[0]; e1 = wrow[1];
    }
    bw[nt].x = e0; bw[nt].y = e1;
    b1v[nt] = (col < HIDDEN) ? b1[col] : 0.f;
  }

  // GEMM2 A tiles: A_kt[m][k] = W2[m][4*kt + k], m<2 valid, rest zero-padded.
  // f32 16x4 A layout: lane m=lid, K = 2*half + vgpr.
  v2f aw2[13];
#pragma unroll
  for (int kt = 0; kt < 13; ++kt) {
    const int k0 = 4 * kt + 2 * half;
    float e0 = 0.f, e1 = 0.f;
    if (lid < 2) {
      if (k0     < HIDDEN) e0 = w2[lid * HIDDEN + k0];
      if (k0 + 1 < HIDDEN) e1 = w2[lid * HIDDEN + k0 + 1];
    }
    aw2[kt].x = e0; aw2[kt].y = e1;
  }

  const float bias0 = b2[0];
  const float bias1 = b2[1];

  // Zero LDS padding columns 50,51 (read by kt=12 but never written by GEMM1 stage).
  Lp[lid * LDS_STRIDE + 50 + half] = 0.f;

  const long total_tiles = ((long)n_rows + 15) / 16;
  const long gwave  = (long)blockIdx.x * WAVES_PER_BLOCK + wave;
  const long nwaves = (long)gridDim.x * WAVES_PER_BLOCK;

  for (long t = gwave; t < total_tiles; t += nwaves) {
    const int row = (int)(t * 16) + lid;

    // A of GEMM1: lanes 0-15 carry x0,x1 (K=0,1); lanes 16-31 zero (K=2,3 pad).
    v2f ax; ax.x = 0.f; ax.y = 0.f;
    if (half == 0 && row < n_rows) {
      const v2f yy = ((const v2f*)y)[row];
      ax.x = yy.x * yy.x * yy.x;
      ax.y = yy.y * yy.y * yy.y;
    }

    // ---- GEMM1: 4 WMMAs over hidden-column tiles ----
    v8f acc[4];
#pragma unroll
    for (int nt = 0; nt < 4; ++nt) {
      v8f z = {};
      acc[nt] = __builtin_amdgcn_wmma_f32_16x16x4_f32(
          false, ax, false, bw[nt], (short)0, z, false, false);
    }

    // ---- tanhshrink + stage to LDS (row = 8*half + r, col = lid + 16*nt) ----
#pragma unroll
    for (int nt = 0; nt < 4; ++nt) {
      const int col = lid + 16 * nt;
      if (col < HIDDEN) {
#pragma unroll
        for (int r = 0; r < 8; ++r) {
          const float h = acc[nt][r] + b1v[nt];
          float th;
          asm volatile("v_tanh_f32 %0, %1\n\tv_nop" : "=v"(th) : "v"(h));
          Lp[(8 * half + r) * LDS_STRIDE + col] = h - th;
        }
      }
    }
    // Same-wave LDS ops are processed in order: no barrier needed (private tile).

    // ---- GEMM2: OUT^T = W2 x A^T, 13 WMMAs, B read transposed from LDS ----
    v8f acc2 = {};
#pragma unroll
    for (int kt = 0; kt < 13; ++kt) {
      // B_kt[k][n] = act[row=n][hid=4*kt+k]; lane: n=lid, K=2*half+vgpr -> b64 load
      const v2f bt = *(const v2f*)&Lp[lid * LDS_STRIDE + 4 * kt + 2 * half];
      acc2 = __builtin_amdgcn_wmma_f32_16x16x4_f32(
          false, aw2[kt], false, bt, (short)0, acc2, false, false);
    }

    // D layout: lane(n=lid), vgpr m: out[row][0]=acc2[0], out[row][1]=acc2[1] in lanes 0-15.
    if (half == 0 && row < n_rows) {
      v2f o; o.x = acc2[0] + bias0; o.y = acc2[1] + bias1;
      ((v2f*)out)[row] = o;
    }
  }
}

extern "C" void kernel_launch(void* const* d_in, const int* in_sizes, int n_in,
                              void* d_out, int out_size, void* d_ws, size_t ws_size,
                              hipStream_t stream) {
  (void)n_in; (void)out_size; (void)d_ws; (void)ws_size;
  // inputs: 0=t(int,unused) 1=y 2=w1 3=b1 4=w2 5=b2 6=saved_param
  const float* y  = (const float*)d_in[1];
  const float* w1 = (const float*)d_in[2];
  const float* b1 = (const float*)d_in[3];
  const float* w2 = (const float*)d_in[4];
  const float* b2 = (const float*)d_in[5];
  const float* sp = (const float*)d_in[6];
  float* out = (float*)d_out;

  const int n_rows = in_sizes[1] / 2;
  const long total_tiles = ((long)n_rows + 15) / 16;
  long maxb = (total_tiles + WAVES_PER_BLOCK - 1) / WAVES_PER_BLOCK;
  int blocks = (int)(maxb < 4096 ? maxb : 4096);
  if (blocks < 1) blocks = 1;

  odefunc_mlp_kernel<<<blocks, 256, 0, stream>>>(y, w1, b1, w2, b2, sp, out, n_rows);
}